// KernelConv2D_46548855554362
// MI455X (gfx1250) — compile-verified
//
#include <hip/hip_runtime.h>

// Per-pixel adaptive (kernel-prediction) 5x5 conv, edge-padded.
// out[b,c,y,x] = sum_{i,j} x_clamped[b,c,y-2+i,x-2+j] * kernel[b,(c*5+i)*5+j,y,x]
//
// Memory-bound: 839 MB kernel stream (read once, NT hint) + 33.5 MB x
// (staged via async global->LDS with halo, reused 25x) + 33.5 MB out (NT store).

typedef float v4f __attribute__((ext_vector_type(4)));

#define B_  4
#define C_  32
#define H_  256
#define W_  256
#define K_  5
#define HALO 2

#define TW 64                 // output tile width  (16 threads * 4 outputs)
#define TH 16                 // output tile height (16 thread rows)
#define LW (TW + 2 * HALO)    // 68 floats per LDS row (272 B, 16B-aligned rows)
#define LH (TH + 2 * HALO)    // 20 rows
#define LDS_ELEMS (LW * LH)   // 1360 floats = 5440 B

__global__ __launch_bounds__(256)
void kpn_conv5x5_kernel(const float* __restrict__ x,
                        const float* __restrict__ kern,
                        float* __restrict__ out)
{
    __shared__ float tile[LDS_ELEMS];

    const int t  = threadIdx.x;        // 0..255
    const int p  = blockIdx.z;         // plane index = b*32 + c, 0..127
    const int x0 = blockIdx.x * TW;
    const int y0 = blockIdx.y * TH;

    const float* xplane = x + (size_t)p * (H_ * W_);

    // ---- Stage x tile (with replication-pad clamping) into LDS via async
    //      global->LDS DMA. Per-lane clamped addresses implement edge padding.
#pragma unroll
    for (int it = 0; it < (LDS_ELEMS + 255) / 256; ++it) {
        int e = t + it * 256;
        e = (e < LDS_ELEMS) ? e : (LDS_ELEMS - 1);   // clamp -> uniform EXEC
        const int r  = e / LW;
        const int cc = e - r * LW;
        int gy = y0 - HALO + r;
        gy = gy < 0 ? 0 : (gy > H_ - 1 ? H_ - 1 : gy);
        int gx = x0 - HALO + cc;
        gx = gx < 0 ? 0 : (gx > W_ - 1 ? W_ - 1 : gx);
        const unsigned voff = (unsigned)((gy * W_ + gx) * 4);      // byte off in plane
        const unsigned lds  = (unsigned)(unsigned long long)&tile[e]; // LDS byte addr
        asm volatile("global_load_async_to_lds_b32 %0, %1, %2"
                     :
                     : "v"(lds), "v"(voff), "s"(xplane)
                     : "memory");
    }
    asm volatile("s_wait_asynccnt 0" ::: "memory");
    __syncthreads();

    // ---- Compute: each thread produces 4 consecutive outputs in one row.
    const int tx = t & 15;
    const int ty = t >> 4;
    const int ox = x0 + tx * 4;
    const int oy = y0 + ty;

    // kernel plane index for tap (i,j) of plane p is 25*p + i*5 + j
    const float* kbase = kern + (size_t)(25 * p) * (H_ * W_)
                              + (size_t)oy * W_ + ox;

    v4f acc = {0.f, 0.f, 0.f, 0.f};

#pragma unroll
    for (int i = 0; i < K_; ++i) {
        // 8 consecutive x values cover taps j=0..4 for 4 outputs (16B aligned)
        const v4f* lrow = (const v4f*)&tile[(ty + i) * LW + tx * 4];
        float xs[8] __attribute__((aligned(16)));
        *(v4f*)&xs[0] = lrow[0];
        *(v4f*)&xs[4] = lrow[1];
#pragma unroll
        for (int j = 0; j < K_; ++j) {
            const float* kp = kbase + (size_t)(i * K_ + j) * (H_ * W_);
            // kernel weights are used exactly once: stream with NT hint so the
            // 839 MB stream does not evict the L2-resident x tensor.
            v4f kv = __builtin_nontemporal_load((const v4f*)kp);
            v4f xv = { xs[j + 0], xs[j + 1], xs[j + 2], xs[j + 3] };
            acc += xv * kv;
        }
    }

    float* op = out + (size_t)p * (H_ * W_) + (size_t)oy * W_ + ox;
    __builtin_nontemporal_store(acc, (v4f*)op);
}

extern "C" void kernel_launch(void* const* d_in, const int* in_sizes, int n_in,
                              void* d_out, int out_size, void* d_ws, size_t ws_size,
                              hipStream_t stream)
{
    (void)in_sizes; (void)n_in; (void)out_size; (void)d_ws; (void)ws_size;
    const float* x    = (const float*)d_in[0];
    const float* kern = (const float*)d_in[1];
    // d_in[2] = kernel_size (5), compile-time constant here.
    float* out = (float*)d_out;

    dim3 grid(W_ / TW, H_ / TH, B_ * C_);   // (4, 16, 128)
    kpn_conv5x5_kernel<<<grid, 256, 0, stream>>>(x, kern, out);
}